// microGIF_15264313770486
// MI455X (gfx1250) — compile-verified
//
#include <hip/hip_runtime.h>

static constexpr int   kN        = 8192;
static constexpr float kThetaInf = 15.0f;
static constexpr float kDeltaU   = 5.0f;
static constexpr float kDDPR     = 3.0f;   // Delta_delay + t_refractory

typedef __attribute__((ext_vector_type(2))) float v2f;
typedef __attribute__((ext_vector_type(8))) float v8f;

// ---------------------------------------------------------------------------
// Prologue: x'[i] = eps(i) * pop_sizes[i] * spiked_prev[i] * neuron_sign[i]
// (row sign folded into the vector so W reduces to raw w + diagonal fixup)
// ---------------------------------------------------------------------------
__global__ __launch_bounds__(256) void gif_prep(
    const float* __restrict__ tss, const float* __restrict__ tau_s,
    const float* __restrict__ pop, const float* __restrict__ spiked_prev,
    const float* __restrict__ sign, float* __restrict__ xprime) {
  int j = blockIdx.x * 256 + threadIdx.x;
  if (j < kN) {
    float t   = tss[j] - kDDPR;
    float eps = (0.5f + 0.5f * tanhf(t)) * expf(-t / tau_s[j]) / tau_s[j];
    xprime[j] = eps * pop[j] * spiked_prev[j] * sign[j];
  }
}

// ---------------------------------------------------------------------------
// Main matvec: partials[split][j] = sum_{i in split} x'[i] * w[i][j]
// 256 threads = 8 wave32; each wave owns 16 columns; 128 columns per block.
// w streamed through LDS with double-buffered async b128 copies; accumulation
// via V_WMMA_F32_16X16X4_F32 with A-rows replicated (answer = row M=0).
// Last stage peeled so the steady-state loop is branch-free:
//   prefetch(s+1) -> s_wait_asynccnt 2 -> barrier -> 4x WMMA -> barrier
// ---------------------------------------------------------------------------
__global__ __launch_bounds__(256) void gif_matvec(
    const float* __restrict__ w, const float* __restrict__ xp,
    float* __restrict__ partials, int rowsPerSplit) {
  __shared__ float tile[2][16 * 128];   // 2 x 8 KB

  const int t    = threadIdx.x;
  const int wv   = t >> 5;        // wave 0..7
  const int lane = t & 31;
  const int col  = lane & 15;     // N within 16-col stripe
  const int hi   = lane >> 4;     // 0: K={0,1}, 1: K={2,3} per fp32 A/B layout
  const int jb0  = blockIdx.x * 128;
  const int ibase   = blockIdx.y * rowsPerSplit;
  const int nstages = rowsPerSplit >> 4;   // 16 rows per stage

  // copy mapping: each thread moves 32 B of the 8 KB stage (2 x b128)
  const int cpyrow = t >> 4;            // 0..15
  const int cpycol = (t & 15) << 3;     // float idx 0..120 step 8

  auto prefetch = [&](int buf, int stage) {
    const float* gsrc =
        w + (long long)(ibase + stage * 16 + cpyrow) * kN + jb0 + cpycol;
    unsigned ldst = (unsigned)(size_t)(&tile[buf][cpyrow * 128 + cpycol]);
    // INST_OFFSET is added to both the LDS and global address (ISA 08 §4.4),
    // so one vdst/vaddr pair covers both 16-byte halves.
    asm volatile(
        "global_load_async_to_lds_b128 %0, %1, off\n\t"
        "global_load_async_to_lds_b128 %0, %1, off offset:16"
        :
        : "v"(ldst), "v"(gsrc)
        : "memory");
  };

  v8f acc = {};

  auto compute = [&](const float* tb, int i0) {
#pragma unroll
    for (int kk = 0; kk < 4; ++kk) {
      const int k0 = kk * 4 + 2 * hi;
      v2f a, b;
      // A (16x4 fp32): rows replicated; VGPR0 = K{0|2}, VGPR1 = K{1|3}
      a.x = xp[i0 + k0];
      a.y = xp[i0 + k0 + 1];
      // B (4x16 fp32): VGPR0 lanes0-15 K=0 / lanes16-31 K=2; VGPR1 K=1/K=3
      b.x = tb[(k0) * 128 + wv * 16 + col];
      b.y = tb[(k0 + 1) * 128 + wv * 16 + col];
      acc = __builtin_amdgcn_wmma_f32_16x16x4_f32(
          false, a, false, b, (short)0, acc, false, false);
    }
  };

  prefetch(0, 0);
  // steady state: branch-free double-buffered pipeline
  for (int s = 0; s < nstages - 1; ++s) {
    prefetch((s + 1) & 1, s + 1);                      // copy s+1 || compute s
    asm volatile("s_wait_asynccnt 0x2" ::: "memory");  // stage s done (in-order)
    __syncthreads();
    compute(&tile[s & 1][0], ibase + s * 16);
    __syncthreads();                                   // buffer s&1 reusable
  }
  // peeled final stage
  asm volatile("s_wait_asynccnt 0x0" ::: "memory");
  __syncthreads();
  compute(&tile[(nstages - 1) & 1][0], ibase + (nstages - 1) * 16);

  // C/D layout: VGPR0, lanes 0-15 hold (M=0, N=lane) -> the matvec result row
  if (lane < 16) {
    partials[(size_t)blockIdx.y * kN + jb0 + wv * 16 + lane] = acc[0];
  }
}

// ---------------------------------------------------------------------------
// Epilogue: reduce splits, diagonal fixup, theta/membrane update, outputs.
// ---------------------------------------------------------------------------
__global__ __launch_bounds__(256) void gif_finish(
    const float* __restrict__ I_ext, const float* __restrict__ noise,
    const float* __restrict__ w, const float* __restrict__ E_L,
    const float* __restrict__ tau_m, const float* __restrict__ tau_theta,
    const float* __restrict__ J_theta, const float* __restrict__ v,
    const float* __restrict__ spiked_prev, const float* __restrict__ theta_v,
    const float* __restrict__ R_m, const float* __restrict__ c,
    const float* __restrict__ xprime, const float* __restrict__ partials,
    int nsplit, float* __restrict__ out) {
  int j = blockIdx.x * 256 + threadIdx.x;
  if (j < kN) {
    float sum = 0.0f;
    for (int s = 0; s < nsplit; ++s) sum += partials[(size_t)s * kN + j];
    sum -= xprime[j] * w[(long long)j * kN + j];      // self-recurrence mask
    float I_syn = tau_m[j] * sum / R_m[j];
    float th    = theta_v[j] +
                  (kThetaInf - theta_v[j] + J_theta[j] * spiked_prev[j]) /
                      tau_theta[j];
    float vn  = v[j] + (E_L[j] - v[j] + R_m[j] * (I_syn + I_ext[j])) / tau_m[j];
    float lam = c[j] * expf((vn - th) / kDeltaU);
    float lamc = fminf(fmaxf(lam, 0.0f), 1.0f);
    out[j]      = lam;
    out[kN + j] = (noise[j] < lamc) ? 1.0f : 0.0f;
  }
}

// ---------------------------------------------------------------------------
extern "C" void kernel_launch(void* const* d_in, const int* in_sizes, int n_in,
                              void* d_out, int out_size, void* d_ws,
                              size_t ws_size, hipStream_t stream) {
  (void)in_sizes; (void)n_in; (void)out_size;
  const float* I_ext       = (const float*)d_in[0];
  const float* noise       = (const float*)d_in[1];
  const float* w           = (const float*)d_in[2];
  const float* E_L         = (const float*)d_in[3];
  const float* tau_m       = (const float*)d_in[4];
  const float* tau_s       = (const float*)d_in[5];
  const float* tau_theta   = (const float*)d_in[6];
  const float* J_theta     = (const float*)d_in[7];
  const float* v           = (const float*)d_in[8];
  const float* spiked_prev = (const float*)d_in[9];
  const float* tss         = (const float*)d_in[10];
  const float* theta_v     = (const float*)d_in[11];
  const float* R_m         = (const float*)d_in[12];
  const float* c           = (const float*)d_in[13];
  const float* pop         = (const float*)d_in[14];
  const float* sign        = (const float*)d_in[15];
  float* out = (float*)d_out;

  // Workspace: [ x'(N) | partials(S x N) ], S = power of two <= 8 that fits.
  float* xprime = (float*)d_ws;
  size_t avail  = ws_size / sizeof(float);
  int S = 8;
  while (S > 1 && (size_t)(S + 1) * kN > avail) S >>= 1;
  float* partials = xprime + kN;

  gif_prep<<<kN / 256, 256, 0, stream>>>(tss, tau_s, pop, spiked_prev, sign,
                                         xprime);
  gif_matvec<<<dim3(kN / 128, S), 256, 0, stream>>>(w, xprime, partials,
                                                    kN / S);
  gif_finish<<<kN / 256, 256, 0, stream>>>(I_ext, noise, w, E_L, tau_m,
                                           tau_theta, J_theta, v, spiked_prev,
                                           theta_v, R_m, c, xprime, partials,
                                           S, out);
}